// CanonicalCorrelationMatcher_65962107732749
// MI455X (gfx1250) — compile-verified
//
#include <hip/hip_runtime.h>
#include <hip/hip_bf16.h>
#include <math.h>

typedef __attribute__((ext_vector_type(2))) float v2f;
typedef __attribute__((ext_vector_type(8))) float v8f;

#define Bq 2
#define Nn 4096
#define Cd 64
#define TOPKk 24
#define RADIUSf 0.45f
// base-2 softmax scale: log2(e)/TEMP
#define INVT2 20.609930f
#define LN2f 0.69314718056f
#define NEGBIG (-1.0e30f)

__device__ __forceinline__ float gridc(int i) { return -1.0f + (2.0f/15.0f)*(float)i; }
__device__ __forceinline__ float fexp2(float x) { return __builtin_amdgcn_exp2f(x); }
__device__ __forceinline__ float flog2(float x) { return __builtin_amdgcn_logf(x); }
__device__ __forceinline__ float fsqrt_fast(float x) { return __builtin_amdgcn_sqrtf(x); }

// ---------------------------------------------------------------------------
// Kernel 1: per-column pack (tc xyz + column score addend) and src matchability
// ---------------------------------------------------------------------------
__global__ void prep_kernel(const float* __restrict__ tgt_c,
                            const float* __restrict__ tml,
                            const float* __restrict__ tunc,
                            const float* __restrict__ sml,
                            float4* __restrict__ colpack,
                            float* __restrict__ smatch) {
    int t = blockIdx.x * blockDim.x + threadIdx.x;
    if (t >= Bq * Nn) return;
    int b = t / Nn, j = t % Nn;
    const float* tcb = tgt_c + (size_t)b * 3 * Nn;
    float4 cp;
    cp.x = tcb[j];
    cp.y = tcb[Nn + j];
    cp.z = tcb[2 * Nn + j];
    // softmax(logits)[...,0] = sigmoid(l0 - l1)
    float l0 = tml[(size_t)b * 2 * Nn + j];
    float l1 = tml[(size_t)b * 2 * Nn + Nn + j];
    float tm = 1.0f / (1.0f + __expf(l1 - l0));
    float lc = fmaxf(__logf(tm), -20.0f);          // log(0) -> -inf -> clamp -20
    cp.w = lc - 0.1f * tunc[(size_t)b * Nn + j];   // column addend
    colpack[t] = cp;
    float s0 = sml[(size_t)b * 2 * Nn + j];
    float s1 = sml[(size_t)b * 2 * Nn + Nn + j];
    smatch[t] = 1.0f / (1.0f + __expf(s1 - s0));
}

// ---------------------------------------------------------------------------
// Kernel 2: per-row 24th-smallest distance via bisection -> allowed threshold
// One 256-thread block per (b, i); each thread keeps 16 dists in registers.
// ---------------------------------------------------------------------------
__global__ __launch_bounds__(256) void thr_kernel(const float* __restrict__ src_c,
                                                  const float4* __restrict__ colpack,
                                                  float* __restrict__ thrw) {
    int blk = blockIdx.x;
    int b = blk / Nn, i = blk % Nn;
    int tid = threadIdx.x;
    const float* scb = src_c + (size_t)b * 3 * Nn;
    float sx = scb[i], sy = scb[Nn + i], sz = scb[2 * Nn + i];
    const float4* cpb = colpack + (size_t)b * Nn;
    float dv[16];
#pragma unroll
    for (int k = 0; k < 16; k++) {
        int j = k * 256 + tid;
        float4 cp = cpb[j];
        float dx = sx - cp.x, dy = sy - cp.y, dz = sz - cp.z;
        dv[k] = fsqrt_fast(fmaxf(dx * dx + dy * dy + dz * dz, 1e-12f));
    }
    __shared__ int cnt;
    float lo = 0.0f, hi = 64.0f;
    for (int it = 0; it < 24; it++) {
        float mid = 0.5f * (lo + hi);
        if (tid == 0) cnt = 0;
        __syncthreads();
        int c = 0;
#pragma unroll
        for (int k = 0; k < 16; k++) c += (dv[k] <= mid) ? 1 : 0;
        atomicAdd(&cnt, c);
        __syncthreads();
        int total = cnt;
        __syncthreads();
        if (total >= TOPKk) hi = mid; else lo = mid;
    }
    if (tid == 0) thrw[blk] = fmaxf(hi, RADIUSf);
}

// ---------------------------------------------------------------------------
// Kernel 3: fused sim(WMMA f32) + dist + mask + online softmax + all outputs.
// Block = 16 rows of one batch; 8 wave32, each wave owns 32 16-col tiles.
// Fully branchless in the WMMA loops (EXEC stays all-ones); base-2 softmax
// with per-row K = M + log2(S) so p = exp2(y - K): no divides, no per-elem log.
// ---------------------------------------------------------------------------
__global__ __launch_bounds__(256) void match_kernel(
    const float* __restrict__ src_c, const float* __restrict__ src_d,
    const float* __restrict__ tgt_d,
    const float4* __restrict__ colpack, const float* __restrict__ smatch,
    const float* __restrict__ thrw, float* __restrict__ out) {

    const int blk = blockIdx.x;
    const int b = blk >> 8;            // 256 row-blocks per batch
    const int r0 = (blk & 255) << 4;   // first of 16 rows
    const int tid = threadIdx.x;
    const int lane = tid & 31;
    const int wave = tid >> 5;
    const int hv = lane >> 4;          // wave half
    const int l15 = lane & 15;

    __shared__ float sc_s[16][3];
    __shared__ float thr_s[16];
    __shared__ float sm_s[16];
    __shared__ float redM[8][16];
    __shared__ float redS[8][16];
    __shared__ float rowK[16];         // M2 + log2(S) per row
    __shared__ float red2[8][16][5];

    if (tid < 16) {
        int r = r0 + tid;
        const float* scb = src_c + (size_t)b * 3 * Nn;
        sc_s[tid][0] = scb[r];
        sc_s[tid][1] = scb[Nn + r];
        sc_s[tid][2] = scb[2 * Nn + r];
        thr_s[tid] = thrw[b * Nn + r];
        sm_s[tid]  = smatch[b * Nn + r];
    }

    // A fragments (16x4 f32 WMMA layout): loaded once, reused 64x.
    v2f afrag[16];
    {
        const float* sdb = src_d + (size_t)b * Cd * Nn + r0 + l15;
#pragma unroll
        for (int kk = 0; kk < 16; kk++) {
            int kr = kk * 4 + 2 * hv;
            afrag[kk].x = sdb[(size_t)kr * Nn];
            afrag[kk].y = sdb[(size_t)(kr + 1) * Nn];
        }
    }
    __syncthreads();

    // Hoist this lane's 8 rows of sc/thr into registers (kills inner-loop LDS)
    float rscx[8], rscy[8], rscz[8], rthr[8];
#pragma unroll
    for (int v = 0; v < 8; v++) {
        int row = v + 8 * hv;
        rscx[v] = sc_s[row][0];
        rscy[v] = sc_s[row][1];
        rscz[v] = sc_s[row][2];
        rthr[v] = thr_s[row];
    }

    const float4* cpb = colpack + (size_t)b * Nn;
    const float* tdb = tgt_d + (size_t)b * Cd * Nn;

    const size_t offDisp  = (size_t)Bq * Nn * 3;
    const size_t offProbs = offDisp + (size_t)Bq * Nn * 3;
    const size_t offConf  = offProbs + (size_t)Bq * Nn * Nn;
    const size_t offEnt   = offConf + (size_t)Bq * Nn;
    const size_t offSrc   = offEnt + (size_t)Bq * Nn;
    float* probs = out + offProbs;

    // ---------------- pass 1: online softmax stats in base-2 ----------------
    float mAcc[8], sAcc[8];
#pragma unroll
    for (int v = 0; v < 8; v++) { mAcc[v] = NEGBIG; sAcc[v] = 0.0f; }

    for (int t = 0; t < 32; t++) {
        int ct = wave + t * 8;
        int col = ct * 16 + l15;
        float4 cp = cpb[col];
        v8f acc = {};
        const float* tdc = tdb + col;
#pragma unroll
        for (int kk = 0; kk < 16; kk++) {
            int kr = kk * 4 + 2 * hv;
            v2f bf;
            bf.x = tdc[(size_t)kr * Nn];
            bf.y = tdc[(size_t)(kr + 1) * Nn];
            acc = __builtin_amdgcn_wmma_f32_16x16x4_f32(
                false, afrag[kk], false, bf, (short)0, acc, false, false);
        }
#pragma unroll
        for (int v = 0; v < 8; v++) {
            float dx = rscx[v] - cp.x, dy = rscy[v] - cp.y, dz = rscz[v] - cp.z;
            float dist = fsqrt_fast(fmaxf(dx * dx + dy * dy + dz * dz, 1e-12f));
            float y = (acc[v] - dist + cp.w) * INVT2;     // base-2 log-prob
            y = (dist <= rthr[v]) ? y : NEGBIG;           // branchless mask
            float m2 = fmaxf(mAcc[v], y);
            sAcc[v] = sAcc[v] * fexp2(mAcc[v] - m2) + fexp2(y - m2);
            mAcc[v] = m2;
        }
    }
    // merge across the 16 lanes of each half (xor butterfly)
#pragma unroll
    for (int v = 0; v < 8; v++) {
        float m = mAcc[v], s = sAcc[v];
#pragma unroll
        for (int off = 1; off < 16; off <<= 1) {
            float om = __shfl_xor(m, off, 32);
            float os = __shfl_xor(s, off, 32);
            float m2 = fmaxf(m, om);
            s = s * fexp2(m - m2) + os * fexp2(om - m2);
            m = m2;
        }
        if (l15 == 0) { int row = v + 8 * hv; redM[wave][row] = m; redS[wave][row] = s; }
    }
    __syncthreads();
    if (tid < 16) {
        float m = NEGBIG, s = 0.0f;
        for (int w = 0; w < 8; w++) {
            float om = redM[w][tid], os = redS[w][tid];
            float m2 = fmaxf(m, om);
            s = s * fexp2(m - m2) + os * fexp2(om - m2);
            m = m2;
        }
        rowK[tid] = m + flog2(s);   // p = exp2(y - K); S >= 1 so log2 is safe
    }
    __syncthreads();

    float rK[8];
#pragma unroll
    for (int v = 0; v < 8; v++) rK[v] = rowK[v + 8 * hv];

    // ---------------- pass 2: probs + expected + conf + entropy -------------
    float ex0[8], ex1[8], ex2[8], pmx[8], ens[8];
#pragma unroll
    for (int v = 0; v < 8; v++) { ex0[v] = ex1[v] = ex2[v] = ens[v] = 0.0f; pmx[v] = 0.0f; }

    for (int t = 0; t < 32; t++) {
        int ct = wave + t * 8;
        int col = ct * 16 + l15;
        float4 cp = cpb[col];
        float t0 = gridc(col >> 8);
        float t1 = gridc((col >> 4) & 15);
        float t2 = gridc(col & 15);
        v8f acc = {};
        const float* tdc = tdb + col;
#pragma unroll
        for (int kk = 0; kk < 16; kk++) {
            int kr = kk * 4 + 2 * hv;
            v2f bf;
            bf.x = tdc[(size_t)kr * Nn];
            bf.y = tdc[(size_t)(kr + 1) * Nn];
            acc = __builtin_amdgcn_wmma_f32_16x16x4_f32(
                false, afrag[kk], false, bf, (short)0, acc, false, false);
        }
#pragma unroll
        for (int v = 0; v < 8; v++) {
            int row = v + 8 * hv;
            float dx = rscx[v] - cp.x, dy = rscy[v] - cp.y, dz = rscz[v] - cp.z;
            float dist = fsqrt_fast(fmaxf(dx * dx + dy * dy + dz * dz, 1e-12f));
            float y = (acc[v] - dist + cp.w) * INVT2;
            y = (dist <= rthr[v]) ? y : NEGBIG;
            float u = y - rK[v];            // base-2 log prob (finite)
            float p = fexp2(u);             // masked -> exact 0
            probs[((size_t)(b * Nn + r0 + row)) * Nn + col] = p;
            ex0[v] += p * t0; ex1[v] += p * t1; ex2[v] += p * t2;
            pmx[v] = fmaxf(pmx[v], p);
            ens[v] += p * u;                // p*log2(p); 0 * finite = 0 when masked
        }
    }
#pragma unroll
    for (int v = 0; v < 8; v++) {
        float a = ex0[v], c = ex1[v], d = ex2[v], e = ens[v], pm = pmx[v];
#pragma unroll
        for (int off = 1; off < 16; off <<= 1) {
            a += __shfl_xor(a, off, 32);
            c += __shfl_xor(c, off, 32);
            d += __shfl_xor(d, off, 32);
            e += __shfl_xor(e, off, 32);
            pm = fmaxf(pm, __shfl_xor(pm, off, 32));
        }
        if (l15 == 0) {
            int row = v + 8 * hv;
            red2[wave][row][0] = a; red2[wave][row][1] = c; red2[wave][row][2] = d;
            red2[wave][row][3] = e; red2[wave][row][4] = pm;
        }
    }
    __syncthreads();
    if (tid < 16) {
        float a = 0, c = 0, d = 0, e = 0, pm = 0;
        for (int w = 0; w < 8; w++) {
            a += red2[w][tid][0]; c += red2[w][tid][1]; d += red2[w][tid][2];
            e += red2[w][tid][3]; pm = fmaxf(pm, red2[w][tid][4]);
        }
        int r = r0 + tid;
        float p0 = gridc(r >> 8), p1 = gridc((r >> 4) & 15), p2 = gridc(r & 15);
        size_t er = (size_t)(b * Nn + r) * 3;
        out[er + 0] = a; out[er + 1] = c; out[er + 2] = d;                 // expected
        out[offDisp + ((size_t)b * 3 + 0) * Nn + r] = a - p0;              // disp
        out[offDisp + ((size_t)b * 3 + 1) * Nn + r] = c - p1;
        out[offDisp + ((size_t)b * 3 + 2) * Nn + r] = d - p2;
        out[offConf + b * Nn + r] = pm * sm_s[tid];                        // conf
        out[offEnt + b * Nn + r] = -e * LN2f;                              // entropy (nats)
        out[offSrc + er + 0] = p0; out[offSrc + er + 1] = p1; out[offSrc + er + 2] = p2;
    }
}

// ---------------------------------------------------------------------------
extern "C" void kernel_launch(void* const* d_in, const int* in_sizes, int n_in,
                              void* d_out, int out_size, void* d_ws, size_t ws_size,
                              hipStream_t stream) {
    const float* src_c = (const float*)d_in[0];
    const float* tgt_c = (const float*)d_in[1];
    const float* src_d = (const float*)d_in[2];
    const float* tgt_d = (const float*)d_in[3];
    const float* sml   = (const float*)d_in[4];
    const float* tml   = (const float*)d_in[5];
    // d_in[6] = src_unc: provably cancels in softmax (row constant; masked
    // entries underflow to exact 0), so unused.
    const float* tunc  = (const float*)d_in[7];
    float* out = (float*)d_out;

    char* ws = (char*)d_ws;
    float4* colpack = (float4*)ws;                        // B*N float4 = 128KB
    float* smatch   = (float*)(ws + 131072);              // 32KB
    float* thrw     = (float*)(ws + 131072 + 32768);      // 32KB

    prep_kernel<<<(Bq * Nn + 255) / 256, 256, 0, stream>>>(tgt_c, tml, tunc, sml, colpack, smatch);
    thr_kernel<<<Bq * Nn, 256, 0, stream>>>(src_c, colpack, thrw);
    match_kernel<<<Bq * (Nn / 16), 256, 0, stream>>>(src_c, src_d, tgt_d, colpack, smatch, thrw, out);
}